// FasterRCNN_8160437862427
// MI455X (gfx1250) — compile-verified
//
#include <hip/hip_runtime.h>
#include <hip/hip_bf16.h>
#include <float.h>

#define TOPK      2000
#define NMS_THR   0.5f
#define CAND_CAP  4096
#define GT_MAX    256

typedef __attribute__((ext_vector_type(2))) float v2f;
typedef __attribute__((ext_vector_type(8))) float v8f;
typedef int i32x4 __attribute__((vector_size(16)));

// ---------------------------------------------------------------------------
// gfx1250 async global->LDS copy helpers (ASYNCcnt path), guarded so compile
// cannot fail if the builtins are absent on this toolchain.
// Builtin signature (from hipcc diagnostic): (i32x4 as1*, i32x4 as3*, Ii, Ii)
// ---------------------------------------------------------------------------
#if __has_builtin(__builtin_amdgcn_global_load_async_to_lds_b128)
#define HAVE_ASYNC_LDS 1
#else
#define HAVE_ASYNC_LDS 0
#endif

__device__ __forceinline__ void async_copy_b128(const void* gsrc, void* ldst) {
#if HAVE_ASYNC_LDS
    __builtin_amdgcn_global_load_async_to_lds_b128(
        (__attribute__((address_space(1))) i32x4*)(uintptr_t)gsrc,
        (__attribute__((address_space(3))) i32x4*)(uintptr_t)ldst,
        0, 0);
#else
    (void)gsrc; (void)ldst;
#endif
}

__device__ __forceinline__ void async_wait0() {
#if HAVE_ASYNC_LDS && __has_builtin(__builtin_amdgcn_s_wait_asynccnt)
    __builtin_amdgcn_s_wait_asynccnt(0);
#endif
}

// ---------------------------------------------------------------------------
// K0: zero histogram / scalar scratch region
// ---------------------------------------------------------------------------
__global__ void zero_u32_kernel(unsigned* p, int n) {
    int i = blockIdx.x * blockDim.x + threadIdx.x;
    if (i < n) p[i] = 0u;
}

// ---------------------------------------------------------------------------
// K1: per-proposal softmax-argmax, single-class decode+clamp, topk key
// ---------------------------------------------------------------------------
__global__ void score_decode_kernel(const float* __restrict__ proposals,
                                    const float* __restrict__ btp,
                                    const float* __restrict__ cls,
                                    int N, int C,
                                    const int* __restrict__ hptr,
                                    const int* __restrict__ wptr,
                                    float* __restrict__ boxPer,
                                    unsigned* __restrict__ keys) {
    int i = blockIdx.x * blockDim.x + threadIdx.x;
    if (i >= N) return;
    // gfx1250 prefetch of the transform row we will gather from later
    __builtin_prefetch(btp + (size_t)i * C * 4, 0, 1);

    const float* s = cls + (size_t)i * C;
    float m = s[0];
    for (int c = 1; c < C; ++c) m = fmaxf(m, s[c]);
    float sum = 0.0f;
    float bestS = -FLT_MAX;
    int   bcls  = 1;
    for (int c = 0; c < C; ++c) {
        float sc = s[c];
        sum += __expf(sc - m);
        if (c >= 1 && sc > bestS) { bestS = sc; bcls = c; }   // first-max tie rule
    }
    float score = __expf(bestS - m) / sum;

    // decode only the winning class
    float4 pb = ((const float4*)proposals)[i];
    float w  = pb.z - pb.x, h = pb.w - pb.y;
    float cx = pb.x + 0.5f * w, cy = pb.y + 0.5f * h;
    float4 t = ((const float4*)btp)[(size_t)i * C + bcls];
    float pcx = t.x * w + cx;
    float pcy = t.y * h + cy;
    float pw  = __expf(t.z) * w;
    float ph  = __expf(t.w) * h;
    float W = (float)wptr[0], H = (float)hptr[0];
    float x1 = fminf(fmaxf(pcx - 0.5f * pw, 0.0f), W);
    float y1 = fminf(fmaxf(pcy - 0.5f * ph, 0.0f), H);
    float x2 = fminf(fmaxf(pcx + 0.5f * pw, 0.0f), W);
    float y2 = fminf(fmaxf(pcy + 0.5f * ph, 0.0f), H);
    ((float4*)boxPer)[i] = make_float4(x1, y1, x2, y2);

    // order-preserving uint key for radix select (score > 0 so MSB always set)
    unsigned u = __float_as_uint(score);
    keys[i] = (u & 0x80000000u) ? ~u : (u | 0x80000000u);
}

// ---------------------------------------------------------------------------
// K2: gt-vs-proposal IoU argmax + box encode.
// 16x16 tiles; the pairwise union term area_gt[m]+area_p[n] is a rank-2
// matmul done with V_WMMA_F32_16X16X4_F32 (exact in f32). gt boxes staged
// into LDS via async global->LDS copies.
// ---------------------------------------------------------------------------
__global__ void match_encode_kernel(const float* __restrict__ proposals,
                                    const float* __restrict__ gt,
                                    int G, int N,
                                    float* __restrict__ outReg) {
    __shared__ float sGt[GT_MAX * 4];
    __shared__ float sArea[GT_MAX];
    int tid = threadIdx.x;
    int Gp  = (G + 15) & ~15;                 // padded tile count (pad = clamp-dup)

#if HAVE_ASYNC_LDS
    for (int g = tid; g < G; g += blockDim.x)
        async_copy_b128(gt + (size_t)g * 4, &sGt[g * 4]);
    async_wait0();
    __syncthreads();
    for (int g = tid; g < Gp; g += blockDim.x) {
        if (g >= G) {
            sGt[g * 4 + 0] = sGt[(G - 1) * 4 + 0];
            sGt[g * 4 + 1] = sGt[(G - 1) * 4 + 1];
            sGt[g * 4 + 2] = sGt[(G - 1) * 4 + 2];
            sGt[g * 4 + 3] = sGt[(G - 1) * 4 + 3];
        }
        sArea[g] = (sGt[g * 4 + 2] - sGt[g * 4 + 0]) *
                   (sGt[g * 4 + 3] - sGt[g * 4 + 1]);
    }
#else
    for (int g = tid; g < Gp; g += blockDim.x) {
        int gg = g < G ? g : (G - 1);
        float4 b = ((const float4*)gt)[gg];
        sGt[g * 4 + 0] = b.x; sGt[g * 4 + 1] = b.y;
        sGt[g * 4 + 2] = b.z; sGt[g * 4 + 3] = b.w;
        sArea[g] = (b.z - b.x) * (b.w - b.y);
    }
#endif
    __syncthreads();

    int lane = tid & 31;
    int hi   = lane >> 4;          // which half-row group of the C tile
    int ln   = lane & 15;          // tile column (proposal)
    long wave  = (long)((blockIdx.x * blockDim.x + tid) >> 5);
    long pbase = wave * 16;
    long pclmp = pbase + ln; if (pclmp > N - 1) pclmp = N - 1;
    float4 pb = ((const float4*)proposals)[pclmp];
    float pw = pb.z - pb.x, ph = pb.w - pb.y;
    float areaP = pw * ph;

    // B (4x16): row0 = ones, row1 = area_p. VGPR0:{row0|row2}, VGPR1:{row1|row3}
    v2f bvec;
    bvec.x = hi ? 0.0f : 1.0f;
    bvec.y = hi ? 0.0f : areaP;

    float bestIou = -1.0f;
    int   bestIdx = 0;
    int nTiles = Gp >> 4;
    for (int t = 0; t < nTiles; ++t) {
        // A (16x4): col0 = area_gt, col1 = ones. VGPR0:{K0|K2}, VGPR1:{K1|K3}
        v2f avec;
        avec.x = hi ? 0.0f : sArea[t * 16 + ln];
        avec.y = hi ? 0.0f : 1.0f;
        v8f dsum;
#if __has_builtin(__builtin_amdgcn_wmma_f32_16x16x4_f32)
        v8f cz = {0.f,0.f,0.f,0.f,0.f,0.f,0.f,0.f};
        dsum = __builtin_amdgcn_wmma_f32_16x16x4_f32(
            false, avec, false, bvec, (short)0, cz, false, false);
#else
        #pragma unroll
        for (int v = 0; v < 8; ++v)
            dsum[v] = sArea[t * 16 + v + 8 * hi] + areaP;
#endif
        #pragma unroll
        for (int v = 0; v < 8; ++v) {
            int m = t * 16 + v + 8 * hi;          // C-layout: M = v + 8*hi
            float gx1 = sGt[m * 4 + 0], gy1 = sGt[m * 4 + 1];
            float gx2 = sGt[m * 4 + 2], gy2 = sGt[m * 4 + 3];
            float iw = fmaxf(0.0f, fminf(gx2, pb.z) - fmaxf(gx1, pb.x));
            float ih = fmaxf(0.0f, fminf(gy2, pb.w) - fmaxf(gy1, pb.y));
            float inter = iw * ih;
            float iou = inter / (dsum[v] - inter);
            if (m < G && iou > bestIou) { bestIou = iou; bestIdx = m; }
        }
    }
    // merge the two half-row groups of this column (lane ^ 16)
    float oIou = __shfl_xor(bestIou, 16, 32);
    int   oIdx = __shfl_xor(bestIdx, 16, 32);
    if (oIou > bestIou || (oIou == bestIou && oIdx < bestIdx)) {
        bestIou = oIou; bestIdx = oIdx;
    }

    long pidx = pbase + ln;
    if (hi == 0 && pidx < N) {
        float cx = pb.x + 0.5f * pw, cy = pb.y + 0.5f * ph;
        float gx1 = sGt[bestIdx * 4 + 0], gy1 = sGt[bestIdx * 4 + 1];
        float gx2 = sGt[bestIdx * 4 + 2], gy2 = sGt[bestIdx * 4 + 3];
        float gw = gx2 - gx1, gh = gy2 - gy1;
        float gcx = gx1 + 0.5f * gw, gcy = gy1 + 0.5f * gh;
        size_t o = (size_t)pidx * 4;
        outReg[o + 0] = (gcx - cx) / pw;
        outReg[o + 1] = (gcy - cy) / ph;
        outReg[o + 2] = __logf(gw / pw);
        outReg[o + 3] = __logf(gh / ph);
    }
}

// ---------------------------------------------------------------------------
// K3/K5: radix histograms (hi 16 bits, then lo 16 bits within chosen bin)
// ---------------------------------------------------------------------------
__global__ void hist_hi_kernel(const unsigned* __restrict__ keys, int N,
                               unsigned* __restrict__ hist) {
    int i = blockIdx.x * blockDim.x + threadIdx.x;
    if (i < N) atomicAdd(&hist[keys[i] >> 16], 1u);
}

__global__ void hist_lo_kernel(const unsigned* __restrict__ keys, int N,
                               const unsigned* __restrict__ scal,
                               unsigned* __restrict__ hist) {
    int i = blockIdx.x * blockDim.x + threadIdx.x;
    if (i < N) {
        unsigned k = keys[i];
        if ((k >> 16) == scal[0]) atomicAdd(&hist[k & 0xFFFFu], 1u);
    }
}

// K4/K6: serial descending scans (single thread; 64K iterations, negligible)
__global__ void scan_hi_kernel(const unsigned* __restrict__ hist,
                               unsigned* __restrict__ scal) {
    unsigned above = 0; unsigned b = 0;
    for (int i = 65535; i >= 0; --i) {
        unsigned c = hist[i];
        if (above + c >= (unsigned)TOPK) { b = (unsigned)i; break; }
        above += c;
    }
    scal[0] = b;          // hi bin of threshold
    scal[1] = above;      // count strictly above this hi bin
}

__global__ void scan_lo_kernel(const unsigned* __restrict__ hist,
                               unsigned* __restrict__ scal) {
    unsigned need = (unsigned)TOPK - scal[1];
    unsigned above = 0; unsigned lo = 0;
    for (int i = 65535; i >= 0; --i) {
        unsigned c = hist[i];
        if (above + c >= need) { lo = (unsigned)i; break; }
        above += c;
    }
    scal[2] = (scal[0] << 16) | lo;   // exact 32-bit threshold key T
}

// K7: compact all keys >= T (count >= TOPK, ~TOPK in practice)
__global__ void compact_kernel(const unsigned* __restrict__ keys, int N,
                               unsigned* __restrict__ scal,
                               unsigned long long* __restrict__ cand) {
    int i = blockIdx.x * blockDim.x + threadIdx.x;
    if (i >= N) return;
    unsigned k = keys[i];
    if (k >= scal[2]) {
        unsigned pos = atomicAdd(&scal[4], 1u);
        if (pos < CAND_CAP)
            cand[pos] = ((unsigned long long)k << 32) | (unsigned)(~(unsigned)i);
    }
}

// K8: bitonic sort of candidates by (key desc, idx asc) -> canonical top-k
__global__ void __launch_bounds__(1024)
sort_cand_kernel(const unsigned long long* __restrict__ cand,
                 const unsigned* __restrict__ scal,
                 unsigned* __restrict__ topIdx) {
    __shared__ unsigned long long sv[CAND_CAP];
    unsigned n = scal[4]; if (n > CAND_CAP) n = CAND_CAP;
    for (int i = threadIdx.x; i < CAND_CAP; i += blockDim.x)
        sv[i] = (i < (int)n) ? cand[i] : 0ull;    // real keys all >= 2^63, pads sink
    __syncthreads();
    for (int k = 2; k <= CAND_CAP; k <<= 1) {
        for (int j = k >> 1; j > 0; j >>= 1) {
            for (int i = threadIdx.x; i < CAND_CAP; i += blockDim.x) {
                int ix = i ^ j;
                if (ix > i) {
                    unsigned long long a = sv[i], b = sv[ix];
                    bool descBlock = ((i & k) == 0);
                    bool doSwap = descBlock ? (a < b) : (a > b);
                    if (doSwap) { sv[i] = b; sv[ix] = a; }
                }
            }
            __syncthreads();
        }
    }
    for (int r = threadIdx.x; r < TOPK; r += blockDim.x)
        topIdx[r] = ~(unsigned)(sv[r] & 0xFFFFFFFFull);
}

// K9: gather the selected boxes
__global__ void gather_kernel(const unsigned* __restrict__ topIdx,
                              const float* __restrict__ boxPer,
                              float* __restrict__ topBoxes) {
    int r = blockIdx.x * blockDim.x + threadIdx.x;
    if (r < TOPK) {
        unsigned idx = topIdx[r];
        ((float4*)topBoxes)[r] = ((const float4*)boxPer)[idx];
    }
}

// K10: greedy NMS entirely in LDS (boxes staged via async copies)
__global__ void __launch_bounds__(1024)
nms_kernel(const float* __restrict__ topBoxes, float* __restrict__ out) {
    __shared__ float4 sb[2048];
    __shared__ int    sup[2048];
    const float4* tb = (const float4*)topBoxes;
#if HAVE_ASYNC_LDS
    for (int i = threadIdx.x; i < TOPK; i += blockDim.x) {
        async_copy_b128(tb + i, &sb[i]);
        sup[i] = 0;
    }
    async_wait0();
#else
    for (int i = threadIdx.x; i < TOPK; i += blockDim.x) {
        sb[i]  = tb[i];
        sup[i] = 0;
    }
#endif
    __syncthreads();
    for (int i = 0; i < TOPK; ++i) {
        if (!sup[i]) {                         // uniform read from LDS
            float4 bi = sb[i];
            float areaI = (bi.z - bi.x) * (bi.w - bi.y);
            for (int j = i + 1 + threadIdx.x; j < TOPK; j += blockDim.x) {
                float4 bj = sb[j];
                float iw = fmaxf(0.0f, fminf(bi.z, bj.z) - fmaxf(bi.x, bj.x));
                float ih = fmaxf(0.0f, fminf(bi.w, bj.w) - fmaxf(bi.y, bj.y));
                float inter = iw * ih;
                float areaJ = (bj.z - bj.x) * (bj.w - bj.y);
                float iou = inter / (areaI + areaJ - inter);
                if (iou > NMS_THR) sup[j] = 1;
            }
        }
        __syncthreads();
    }
    for (int i = threadIdx.x; i < TOPK; i += blockDim.x) {
        float k = sup[i] ? 0.0f : 1.0f;
        float4 b = sb[i];
        ((float4*)out)[i] = make_float4(b.x * k, b.y * k, b.z * k, b.w * k);
    }
}

// ---------------------------------------------------------------------------
extern "C" void kernel_launch(void* const* d_in, const int* in_sizes, int n_in,
                              void* d_out, int out_size, void* d_ws, size_t ws_size,
                              hipStream_t stream) {
    const float* proposals = (const float*)d_in[0];
    const float* btp       = (const float*)d_in[1];
    const float* cls       = (const float*)d_in[2];
    const float* gt        = (const float*)d_in[3];
    const int*   hptr      = (const int*)d_in[4];
    const int*   wptr      = (const int*)d_in[5];

    const int N = in_sizes[0] / 4;
    const int C = in_sizes[2] / N;
    const int G = in_sizes[3] / 4;

    // workspace layout (all 16B aligned)
    char* ws = (char*)d_ws;
    float*              boxPer   = (float*)ws;               ws += (size_t)N * 16;
    unsigned*           keys     = (unsigned*)ws;            ws += (size_t)N * 4;
    unsigned*           hist1    = (unsigned*)ws;            ws += 65536 * 4;
    unsigned*           hist2    = (unsigned*)ws;            ws += 65536 * 4;
    unsigned*           scal     = (unsigned*)ws;            ws += 64;
    unsigned long long* cand     = (unsigned long long*)ws;  ws += CAND_CAP * 8;
    unsigned*           topIdx   = (unsigned*)ws;            ws += TOPK * 4;
    float*              topBoxes = (float*)ws;               ws += TOPK * 16;

    float* outFinal = (float*)d_out;            // [TOPK,4]
    float* outReg   = (float*)d_out + TOPK * 4; // [N,4]

    // zero histograms + scalars (hist1, hist2, scal are contiguous)
    const int zcnt = 65536 * 2 + 16;
    zero_u32_kernel<<<(zcnt + 255) / 256, 256, 0, stream>>>(hist1, zcnt);

    score_decode_kernel<<<(N + 255) / 256, 256, 0, stream>>>(
        proposals, btp, cls, N, C, hptr, wptr, boxPer, keys);

    const int waves  = (N + 15) / 16;
    const int mblks  = (waves + 7) / 8;        // 256 thr = 8 waves, 16 proposals each
    match_encode_kernel<<<mblks, 256, 0, stream>>>(proposals, gt, G, N, outReg);

    hist_hi_kernel<<<(N + 255) / 256, 256, 0, stream>>>(keys, N, hist1);
    scan_hi_kernel<<<1, 1, 0, stream>>>(hist1, scal);
    hist_lo_kernel<<<(N + 255) / 256, 256, 0, stream>>>(keys, N, scal, hist2);
    scan_lo_kernel<<<1, 1, 0, stream>>>(hist2, scal);
    compact_kernel<<<(N + 255) / 256, 256, 0, stream>>>(keys, N, scal, cand);
    sort_cand_kernel<<<1, 1024, 0, stream>>>(cand, scal, topIdx);
    gather_kernel<<<(TOPK + 255) / 256, 256, 0, stream>>>(topIdx, boxPer, topBoxes);
    nms_kernel<<<1, 1024, 0, stream>>>(topBoxes, outFinal);

    (void)n_in; (void)out_size; (void)ws_size;
}